// GetDensity_19516331393548
// MI455X (gfx1250) — compile-verified
//
#include <hip/hip_runtime.h>
#include <math.h>

// ---------------------------------------------------------------------------
// GetDensity (REANN-style) for MI455X / gfx1250, wave32.
//   - edge geometry + radial/angular basis kept resident in L2 (~15 MB)
//   - orbital recursion folded into per-atom accumulator (avoids [E,13,16])
//   - scatter via hardware global_atomic_add_f32 (unsafeAtomicAdd)
//   - einsum + MLP GEMMs via V_WMMA_F32_16X16X4_F32 (native f32 WMMA)
// ---------------------------------------------------------------------------

typedef float v2f __attribute__((ext_vector_type(2)));
typedef float v8f __attribute__((ext_vector_type(8)));

#define NPARA 13
#define NW    16
#define SLOT  (NPARA * NW)   // 208 floats per atom
#define NORB  128
#define HID   64
#define PI_OVER_CUTOFF 0.62831853071795864769f  // pi / 5.0

// ---------------- prep: ef_orb0 per batch, orb_coeff init --------------------
__global__ void prep_kernel(const float* __restrict__ ef,
                            const float* __restrict__ ef_para,
                            const float* __restrict__ params_p,
                            const int*   __restrict__ species,
                            float* __restrict__ ef0,        // [B, 13, 16]
                            float* __restrict__ orb_coeff,  // [totnatom, 16]
                            int B, int totnatom) {
  int idx = blockIdx.x * blockDim.x + threadIdx.x;
  if (idx < totnatom * NW) {
    int a = idx / NW, k = idx % NW;
    orb_coeff[idx] = params_p[species[a] * NW + k];
  }
  if (idx < B * SLOT) {
    int b = idx / SLOT, s = idx % SLOT;
    int p = s / NW, k = s % NW;
    const float* e = ef + b * 3;
    float av;
    if (p == 0)      av = 1.0f;
    else if (p < 4)  av = e[p - 1];
    else { int q = p - 4; av = e[q / 3] * e[q % 3]; }
    ef0[idx] = av * ef_para[k];
  }
}

// ---------------- per-edge geometry: ang[13], rad[16], dcut ------------------
__global__ void geom_kernel(const float* __restrict__ cart,
                            const float* __restrict__ shifts,
                            const float* __restrict__ rs,
                            const float* __restrict__ inta,
                            const int*   __restrict__ neigh,  // [2, E]
                            const int*   __restrict__ species,
                            float* __restrict__ dcut,         // [E]
                            float* __restrict__ ang,          // [E, 13]
                            float* __restrict__ rad,          // [E, 16]
                            int E) {
  int e = blockIdx.x * blockDim.x + threadIdx.x;
  if (e >= E) return;
  int i = neigh[e], j = neigh[E + e];
  float dx = cart[i * 3 + 0] - cart[j * 3 + 0] - shifts[e * 3 + 0];
  float dy = cart[i * 3 + 1] - cart[j * 3 + 1] - shifts[e * 3 + 1];
  float dz = cart[i * 3 + 2] - cart[j * 3 + 2] - shifts[e * 3 + 2];
  float d   = sqrtf(dx * dx + dy * dy + dz * dz);
  float inv = 1.0f / d;
  float u0 = dx * inv, u1 = dy * inv, u2 = dz * inv;
  float c  = 0.5f * __cosf(d * PI_OVER_CUTOFF) + 0.5f;
  float dc = c * c;
  dcut[e] = dc;
  float* ae = ang + (size_t)e * NPARA;
  ae[0] = dc;
  ae[1] = dc * u0; ae[2] = dc * u1; ae[3] = dc * u2;
  float uu[3] = {u0, u1, u2};
  #pragma unroll
  for (int a = 0; a < 3; a++)
    #pragma unroll
    for (int b = 0; b < 3; b++)
      ae[4 + a * 3 + b] = dc * uu[a] * uu[b];
  int sj = species[j];
  float* re = rad + (size_t)e * NW;
  #pragma unroll
  for (int k = 0; k < NW; k++) {
    float t = d - rs[sj * NW + k];
    re[k] = __expf(inta[sj * NW + k] * t * t);
  }
}

// ------- advance: acc += ef_orb (history), ef_orb <- ef_orb0 (reset) ---------
__global__ void advance_kernel(float* __restrict__ acc,
                               float* __restrict__ ef_orb,
                               const float* __restrict__ ef0,
                               int totnatom, int A, int first) {
  int idx = blockIdx.x * blockDim.x + threadIdx.x;
  if (idx >= totnatom * SLOT) return;
  int a = idx / SLOT, s = idx % SLOT;
  float base = ef0[(a / A) * SLOT + s];
  if (first) acc[idx] = 0.0f;
  else       acc[idx] += ef_orb[idx];
  ef_orb[idx] = base;
}

// ------- scatter: ef_orb[i] += (ang*rad + dcut*acc[j]) * orb_coeff[j] --------
__global__ void scatter_kernel(const float* __restrict__ ang,
                               const float* __restrict__ rad,
                               const float* __restrict__ dcut,
                               const float* __restrict__ acc,
                               const float* __restrict__ orb_coeff,
                               const int*   __restrict__ neigh,
                               float* __restrict__ ef_orb,
                               int E) {
  int idx = blockIdx.x * blockDim.x + threadIdx.x;
  if (idx >= E * NW) return;
  int e = idx >> 4, k = idx & 15;
  int i = neigh[e], j = neigh[E + e];
  float r  = rad[(size_t)e * NW + k];
  float oc = orb_coeff[(size_t)j * NW + k];
  float dc = dcut[e];
  const float* ae = ang + (size_t)e * NPARA;
  const float* aj = acc + (size_t)j * SLOT + k;
  float* ei = ef_orb + (size_t)i * SLOT + k;
  #pragma unroll
  for (int p = 0; p < NPARA; p++) {
    float worb = (ae[p] * r + dc * aj[p * NW]) * oc;
    unsafeAtomicAdd(ei + p * NW, worb);
  }
}

// ------- density: dens[a,m] = sum_p (ef_orb[a,p,:] @ hyper[g(p)][:,m])^2 -----
// 512 blocks (16-atom tiles) x 8 waves (16-col tiles of NORB=128).
// V_WMMA_F32_16X16X4_F32:  A 16x4 (2 VGPR: lanes0-15 K=kb,kb+1; lanes16-31
// K=kb+2,kb+3), B 4x16 (same K split), D 16x16 (VGPR v: M=v / M=v+8).
__global__ __launch_bounds__(256) void density_kernel(
    const float* __restrict__ ef_orb,   // [totnatom, 13, 16]
    const float* __restrict__ hyper,    // [3, 16, 128]
    float* __restrict__ dens) {         // [totnatom, 128]
  int lane = threadIdx.x & 31;
  int wave = threadIdx.x >> 5;          // 0..7 -> orbit-column tile
  int lm = lane & 15, hi = lane >> 4;
  int a0 = blockIdx.x * 16;
  int m0 = wave * 16;
  const float* Arow = ef_orb + (size_t)(a0 + lm) * SLOT;
  v8f acc = {};
  #pragma unroll
  for (int p = 0; p < NPARA; p++) {
    int g = (p == 0) ? 0 : ((p < 4) ? 1 : 2);
    const float* Hp = hyper + (size_t)g * NW * NORB + m0 + lm;
    const float* Ap = Arow + p * NW;
    v8f c = {};
    #pragma unroll
    for (int kb = 0; kb < NW; kb += 4) {
      int ka = kb + 2 * hi;
      v2f Av; Av.x = Ap[ka];          Av.y = Ap[ka + 1];
      v2f Bv; Bv.x = Hp[ka * NORB];   Bv.y = Hp[(ka + 1) * NORB];
      c = __builtin_amdgcn_wmma_f32_16x16x4_f32(
              false, Av, false, Bv, (short)0, c, false, false);
    }
    acc += c * c;                     // square-accumulate hw^2 per p
  }
  float* out = dens + (size_t)a0 * NORB + m0 + lm;
  #pragma unroll
  for (int v = 0; v < 8; v++) out[(v + hi * 8) * NORB] = acc[v];
}

// ------- MLP: orb_coeff += tanh(dens @ w1 + b1) @ w2 -------------------------
// 512 blocks x 4 waves. GEMM1 (K=128) -> LDS h[16,64] -> GEMM2 (K=64, wave 0).
__global__ __launch_bounds__(128) void mlp_kernel(
    const float* __restrict__ dens,     // [totnatom, 128]
    const float* __restrict__ w1,       // [128, 64]
    const float* __restrict__ b1,       // [64]
    const float* __restrict__ w2,       // [64, 16]
    float* __restrict__ orb_coeff) {    // [totnatom, 16]
  __shared__ float hsm[16 * HID];
  int lane = threadIdx.x & 31;
  int wave = threadIdx.x >> 5;          // 0..3 -> HID tile
  int lm = lane & 15, hi = lane >> 4;
  int a0 = blockIdx.x * 16;
  int n0 = wave * 16;
  const float* Arow = dens + (size_t)(a0 + lm) * NORB;
  v8f c = {};
  #pragma unroll 4
  for (int kb = 0; kb < NORB; kb += 4) {
    int ka = kb + 2 * hi;
    v2f Av; Av.x = Arow[ka];                 Av.y = Arow[ka + 1];
    v2f Bv; Bv.x = w1[ka * HID + n0 + lm];   Bv.y = w1[(ka + 1) * HID + n0 + lm];
    c = __builtin_amdgcn_wmma_f32_16x16x4_f32(
            false, Av, false, Bv, (short)0, c, false, false);
  }
  #pragma unroll
  for (int v = 0; v < 8; v++) {
    int row = v + hi * 8, col = n0 + lm;
    hsm[row * HID + col] = tanhf(c[v] + b1[col]);
  }
  __syncthreads();
  if (wave == 0) {
    v8f d = {};
    #pragma unroll
    for (int kb = 0; kb < HID; kb += 4) {
      int ka = kb + 2 * hi;
      v2f Av; Av.x = hsm[lm * HID + ka];   Av.y = hsm[lm * HID + ka + 1];
      v2f Bv; Bv.x = w2[ka * NW + lm];     Bv.y = w2[(ka + 1) * NW + lm];
      d = __builtin_amdgcn_wmma_f32_16x16x4_f32(
              false, Av, false, Bv, (short)0, d, false, false);
    }
    #pragma unroll
    for (int v = 0; v < 8; v++)
      orb_coeff[(size_t)(a0 + v + hi * 8) * NW + lm] += d[v];
  }
}

// ---------------------------------------------------------------------------
extern "C" void kernel_launch(void* const* d_in, const int* in_sizes, int n_in,
                              void* d_out, int out_size, void* d_ws, size_t ws_size,
                              hipStream_t stream) {
  const float* cart     = (const float*)d_in[0];
  const float* ef       = (const float*)d_in[1];
  const float* shifts   = (const float*)d_in[2];
  const float* rs       = (const float*)d_in[3];
  const float* inta     = (const float*)d_in[4];
  const float* params_p = (const float*)d_in[5];
  const float* ef_para  = (const float*)d_in[6];
  const float* hyper    = (const float*)d_in[7];
  const float* oc_w1    = (const float*)d_in[8];
  const float* oc_b1    = (const float*)d_in[9];
  const float* oc_w2    = (const float*)d_in[10];
  const int*   neigh    = (const int*)d_in[11];
  const int*   species  = (const int*)d_in[12];
  float*       dens     = (float*)d_out;

  const int totnatom = in_sizes[0] / 3;        // B*A
  const int B        = in_sizes[1] / 3;
  const int A        = totnatom / B;
  const int E        = in_sizes[2] / 3;
  const int NOC      = in_sizes[9] / HID;      // oc_b1 = [NOC, HID]

  // workspace layout (floats)
  float* ws = (float*)d_ws;
  size_t o = 0;
  float* ef0       = ws + o; o += ((size_t)B * SLOT + 255) & ~(size_t)255;
  float* orb_coeff = ws + o; o += (size_t)totnatom * NW;
  float* dcut      = ws + o; o += (size_t)E;
  float* ang       = ws + o; o += (size_t)E * NPARA;
  float* rad       = ws + o; o += (size_t)E * NW;
  float* ef_orb    = ws + o; o += (size_t)totnatom * SLOT;
  float* acc       = ws + o; o += (size_t)totnatom * SLOT;

  const int TB = 256;
  int n_prep = totnatom * NW;
  if (B * SLOT > n_prep) n_prep = B * SLOT;
  prep_kernel<<<(n_prep + TB - 1) / TB, TB, 0, stream>>>(
      ef, ef_para, params_p, species, ef0, orb_coeff, B, totnatom);

  geom_kernel<<<(E + TB - 1) / TB, TB, 0, stream>>>(
      cart, shifts, rs, inta, neigh, species, dcut, ang, rad, E);

  const int nslot    = totnatom * SLOT;
  const int nscatter = E * NW;
  const int ntiles   = totnatom / 16;

  for (int t = 0; t <= NOC; ++t) {
    advance_kernel<<<(nslot + TB - 1) / TB, TB, 0, stream>>>(
        acc, ef_orb, ef0, totnatom, A, t == 0 ? 1 : 0);
    scatter_kernel<<<(nscatter + TB - 1) / TB, TB, 0, stream>>>(
        ang, rad, dcut, acc, orb_coeff, neigh, ef_orb, E);
    density_kernel<<<ntiles, 256, 0, stream>>>(ef_orb, hyper, dens);
    if (t < NOC) {
      mlp_kernel<<<ntiles, 128, 0, stream>>>(
          dens, oc_w1 + (size_t)t * NORB * HID, oc_b1 + (size_t)t * HID,
          oc_w2 + (size_t)t * HID * NW, orb_coeff);
    }
  }
}